// MLPPixelShifter_26233660244228
// MI455X (gfx1250) — compile-verified
//
#include <hip/hip_runtime.h>
#include <math.h>

typedef __attribute__((ext_vector_type(2))) float        v2f;
typedef __attribute__((ext_vector_type(8))) float        v8f;
typedef __attribute__((ext_vector_type(4))) unsigned int v4u;
typedef __attribute__((ext_vector_type(8))) int          v8i;
typedef __attribute__((ext_vector_type(4))) int          v4i;

#define N_FRAMES 8000
#define HID      100
#define GR       25
#define GSZ      51
#define NPIX     (GSZ * GSZ)
#define IMH      100
#define IMW      100
#define WINMAX   52   // 52 columns/rows can be touched per frame

__device__ __forceinline__ float gelu_exact(float x) {
    // torch nn.GELU default (exact): 0.5*x*(1+erf(x/sqrt(2)))
    return 0.5f * x * (1.0f + erff(x * 0.70710678118654752440f));
}

// ---------------------------------------------------------------------------
// Kernel 1: anchored MLP shift via V_WMMA_F32_16X16X4_F32.
// One wave handles 16 frames. Layer 1 = [16x2]@[2x100] done as 7 WMMA tiles
// (K padded 2->4, N padded 100->112 with zero weights). Layer 2 (N=2) done as
// per-lane products + half-wave shfl_xor reduction. b2 cancels in anchoring.
// ---------------------------------------------------------------------------
__global__ __launch_bounds__(32)
void mlp_shift_wmma(const float* __restrict__ eyepos,
                    const float* __restrict__ W1,
                    const float* __restrict__ b1,
                    const float* __restrict__ W2,
                    float* __restrict__ shift_ws,
                    float* __restrict__ shift_out)
{
    const int lane = threadIdx.x;      // wave32
    const int base = blockIdx.x * 16;  // first frame of this tile
    const int row  = lane & 15;
    const int hi   = lane >> 4;        // 0: holds K=0/1, 1: holds K=2/3 (A); K=2/3 rows of B

    // A matrix 16x4 f32 layout: lanes 0-15 {v0=K0,v1=K1}, lanes 16-31 {v0=K2,v1=K3}
    v2f a;
    a.x = (hi == 0) ? eyepos[(base + row) * 2 + 0] : 0.0f;
    a.y = (hi == 0) ? eyepos[(base + row) * 2 + 1] : 0.0f;

    float accX[8], accY[8];
#pragma unroll
    for (int r = 0; r < 8; ++r) { accX[r] = 0.0f; accY[r] = 0.0f; }

#pragma unroll
    for (int n = 0; n < 7; ++n) {           // hidden columns 16n .. 16n+15
        const int  j  = n * 16 + row;
        const bool jv = (j < HID);

        // B matrix 4x16 f32 layout: v0 = rows K0|K2, v1 = rows K1|K3 (cols = lane&15)
        v2f b;
        b.x = (hi == 0 && jv) ? W1[j * 2 + 0] : 0.0f;
        b.y = (hi == 0 && jv) ? W1[j * 2 + 1] : 0.0f;

        const float bias = jv ? b1[j] : 0.0f;   // same column for both half-waves
        v8f c;
#pragma unroll
        for (int r = 0; r < 8; ++r) c[r] = bias;   // C[m][j] = b1[j] for all m

        // D = A*B + C  ->  layer-1 pre-activation for 16 frames x 16 hidden units
        c = __builtin_amdgcn_wmma_f32_16x16x4_f32(false, a, false, b,
                                                  (short)0, c, false, false);

        const float h0  = gelu_exact(bias);            // mlp(0) hidden (anchor)
        const float w2x = jv ? W2[0 * HID + j] : 0.0f;
        const float w2y = jv ? W2[1 * HID + j] : 0.0f;
#pragma unroll
        for (int r = 0; r < 8; ++r) {
            const float d = gelu_exact(c[r]) - h0;     // anchored activation
            accX[r] += d * w2x;
            accY[r] += d * w2y;
        }
    }

    // Reduce the 16 lanes of each half-wave: lane group (0-15) -> m=r, (16-31) -> m=r+8
#pragma unroll
    for (int r = 0; r < 8; ++r) {
        float sx = accX[r], sy = accY[r];
#pragma unroll
        for (int m = 1; m <= 8; m <<= 1) {
            sx += __shfl_xor(sx, m, 32);
            sy += __shfl_xor(sy, m, 32);
        }
        if (row == 0) {
            const int frame = base + r + hi * 8;
            shift_ws[frame * 2 + 0]  = sx;
            shift_ws[frame * 2 + 1]  = sy;
            shift_out[frame * 2 + 0] = sx;
            shift_out[frame * 2 + 1] = sy;
        }
    }
}

// ---------------------------------------------------------------------------
// Kernel 2: per-frame bilinear sampling. TDM stages the (clamped) window of
// stim[t] that this frame's shifted grid can touch into LDS, then 256 threads
// interpolate 51x51 samples out of LDS with zero padding handled in registers.
// ---------------------------------------------------------------------------
__global__ __launch_bounds__(256)
void bilinear_shift_tdm(const float* __restrict__ stim,
                        const float* __restrict__ shift_ws,
                        float* __restrict__ out)
{
    __shared__ float tile[WINMAX * WINMAX];   // 10.8 KB of the 320 KB WGP LDS

    const int   t  = blockIdx.x;
    const float sx = shift_ws[t * 2 + 0];
    const float sy = shift_ws[t * 2 + 1];
    const int   fx = (int)floorf(sx);
    const int   fy = (int)floorf(sy);

    // Columns/rows reachable by the bilinear taps: [fx-25, fx+26] x [fy-25, fy+26]
    int x_lo = fx - GR;        int x_hi = fx + GR + 1;
    int y_lo = fy - GR;        int y_hi = fy + GR + 1;
    x_lo = x_lo < 0 ? 0 : x_lo;            x_hi = x_hi > IMW - 1 ? IMW - 1 : x_hi;
    y_lo = y_lo < 0 ? 0 : y_lo;            y_hi = y_hi > IMH - 1 ? IMH - 1 : y_hi;
    const int tw = x_hi - x_lo + 1;
    const int th = y_hi - y_lo + 1;

    if (threadIdx.x < 32 && tw > 0 && th > 0) {
        const unsigned lds_base =
            (unsigned)(unsigned long long)(void*)&tile[0];          // LDS byte offset
        const unsigned long long gaddr = (unsigned long long)(const void*)
            (stim + ((size_t)t * (IMH * IMW) + (size_t)y_lo * IMW + x_lo));

        // ---- Tensor DMA descriptor (D#) -------------------------------------
        v4u g0;
        g0.x = 1u;                                        // count=1 (valid user D#)
        g0.y = lds_base;                                  // lds_addr
        g0.z = (unsigned)(gaddr & 0xFFFFFFFFull);         // global_addr[31:0]
        g0.w = (unsigned)((gaddr >> 32) & 0x01FFFFFFull)  // global_addr[56:32]
             | (2u << 30);                                // type = 2 ("image")

        v8i g1;
        g1[0] = (int)(2u << 16);                // data_size = 2 -> 4 bytes/elem
        g1[1] = (int)((unsigned)tw << 16);      // tensor_dim0[15:0]   @bits[63:48]
        g1[2] = (int)((unsigned)th << 16);      // tensor_dim1[15:0]   @bits[95:80]
        g1[3] = (int)((unsigned)tw << 16);      // tile_dim0           @bits[127:112]
        g1[4] = (int)(unsigned)th;              // tile_dim1           @bits[143:128]
        g1[5] = IMW;                            // tensor_dim0_stride  (100 elems)
        g1[6] = 0;
        g1[7] = 0;

        v4i g2 = {0, 0, 0, 0};                  // 2D tensor: groups 2/3 unused
        v4i g3 = {0, 0, 0, 0};
        v8i g4 = {0, 0, 0, 0, 0, 0, 0, 0};      // extra group (6-arg toolchain form)

        __builtin_amdgcn_tensor_load_to_lds(g0, g1, g2, g3, g4, 0);
        __builtin_amdgcn_s_wait_tensorcnt(0);
    }
    __syncthreads();   // publish TDM-written LDS to all 8 waves

    for (int i = threadIdx.x; i < NPIX; i += blockDim.x) {
        const int gy = i / GSZ;
        const int gx = i - gy * GSZ;
        const float xs = (float)(gx - GR) + sx;
        const float ys = (float)(gy - GR) + sy;
        const float xf = floorf(xs), yf = floorf(ys);
        const int   x0 = (int)xf,    y0 = (int)yf;
        const float wx = xs - xf,    wy = ys - yf;

        // Any in-bounds tap is provably inside [x_lo..x_hi] x [y_lo..y_hi]
        auto fetch = [&](int xi, int yi) -> float {
            if (xi < 0 || xi >= IMW || yi < 0 || yi >= IMH) return 0.0f;
            return tile[(yi - y_lo) * tw + (xi - x_lo)];
        };
        const float v00 = fetch(x0,     y0);
        const float v01 = fetch(x0 + 1, y0);
        const float v10 = fetch(x0,     y0 + 1);
        const float v11 = fetch(x0 + 1, y0 + 1);

        out[(size_t)t * NPIX + i] =
            v00 * (1.0f - wx) * (1.0f - wy) + v01 * wx * (1.0f - wy) +
            v10 * (1.0f - wx) * wy          + v11 * wx * wy;
    }
}

// ---------------------------------------------------------------------------
// inputs (setup_inputs order): stim, eyepos, grid, W1, b1, W2, b2
// d_out: [8000*51*51] out  ++  [8000*2] shift_out   (float32)
// ---------------------------------------------------------------------------
extern "C" void kernel_launch(void* const* d_in, const int* in_sizes, int n_in,
                              void* d_out, int out_size, void* d_ws, size_t ws_size,
                              hipStream_t stream) {
    (void)in_sizes; (void)n_in; (void)out_size; (void)ws_size;
    const float* stim   = (const float*)d_in[0];
    const float* eyepos = (const float*)d_in[1];
    // d_in[2] = grid: implicit (integer lattice -25..25), not needed
    const float* W1     = (const float*)d_in[3];
    const float* b1     = (const float*)d_in[4];
    const float* W2     = (const float*)d_in[5];
    // d_in[6] = b2: cancels under anchoring

    float* out       = (float*)d_out;
    float* shift_out = out + (size_t)N_FRAMES * NPIX;
    float* shift_ws  = (float*)d_ws;   // 8000*2 floats of scratch

    mlp_shift_wmma<<<N_FRAMES / 16, 32, 0, stream>>>(eyepos, W1, b1, W2,
                                                     shift_ws, shift_out);
    bilinear_shift_tdm<<<N_FRAMES, 256, 0, stream>>>(stim, shift_ws, out);
}